// GraphPropagation_5772436046227
// MI455X (gfx1250) — compile-verified
//
#include <hip/hip_runtime.h>
#include <hip/hip_bf16.h>
#include <math.h>

// Problem constants (from reference setup_inputs)
#define KDIM 6
#define BDIM 128
#define NDIM 65536
#define DDIM 256
#define TOPK 5
#define CHUNK_K 64
#define INV_TEMP (1.0f / 3.0f)

typedef __attribute__((ext_vector_type(2))) float v2f;
typedef __attribute__((ext_vector_type(8))) float v8f;

// ---------------------------------------------------------------------------
// Kernel 1: row-wise L2 normalization of part_features [K*B][D] -> feats
// ---------------------------------------------------------------------------
__global__ void __launch_bounds__(DDIM)
normalize_kernel(const float* __restrict__ pf, float* __restrict__ feats) {
    const int row = blockIdx.x;          // 0 .. K*B-1
    const int tid = threadIdx.x;         // 0 .. 255 (== D)
    const size_t off = (size_t)row * DDIM + tid;
    const float x = pf[off];

    __shared__ float red[DDIM];
    red[tid] = x * x;
    __syncthreads();
    for (int s = DDIM / 2; s > 0; s >>= 1) {
        if (tid < s) red[tid] += red[tid + s];
        __syncthreads();
    }
    float nrm = fmaxf(sqrtf(red[0]), 1e-12f);
    feats[off] = x / nrm;
}

// ---------------------------------------------------------------------------
// Kernel 2: sim[k] = feats[k] @ memory_bank[k]^T  via V_WMMA_F32_16X16X4_F32
//   block = 128 threads (4 waves). blockIdx.y = k, blockIdx.x = 128-col group.
//   Each wave: TWO 16-col tiles (32 cols), all 128 b-rows
//   -> 16 x v8f accumulators; 8 LDS A-fragments feed 16 WMMAs per K-step.
//   A staged in LDS in K-chunks of 64 (32 KB), shared by the 4 waves.
//   Also zero-fills the soft_labels half of d_out at the same indices.
// ---------------------------------------------------------------------------
__global__ void __launch_bounds__(128)
sim_gemm_kernel(const float* __restrict__ feats,  // [K][B][D]
                const float* __restrict__ mem,    // [K][N][D]
                float* __restrict__ sim,          // [K][B][N]
                float* __restrict__ soft) {       // [K][B][N] (zero-filled here)
    extern __shared__ float ldsA[];               // [BDIM][CHUNK_K] = 32 KB

    const int k    = blockIdx.y;
    const int wave = threadIdx.x >> 5;            // 0..3
    const int lane = threadIdx.x & 31;
    const int l16  = lane & 15;
    const int half = lane >> 4;                   // 0: K=0,1  1: K=2,3
    const int n0   = blockIdx.x * 128 + wave * 32;   // this wave's 32 columns

    const float* fk = feats + (size_t)k * BDIM * DDIM;
    const float* mk = mem   + (size_t)k * NDIM * DDIM;

    v8f acc0[8], acc1[8];
#pragma unroll
    for (int t = 0; t < 8; ++t) { acc0[t] = (v8f)0.0f; acc1[t] = (v8f)0.0f; }

    // B-fragment bases: lane -> memory_bank rows (n0+l16) and (n0+16+l16)
    const float* brow0 = mk + (size_t)(n0 + l16) * DDIM + 2 * half;
    const float* brow1 = brow0 + (size_t)16 * DDIM;

    for (int kc = 0; kc < DDIM; kc += CHUNK_K) {
        __syncthreads();
        // cooperative load of A chunk: 128x64 floats, 128 threads x 16 float4
        for (int i = threadIdx.x; i < BDIM * CHUNK_K / 4; i += 128) {
            const int flat = i * 4;
            const int r = flat / CHUNK_K;
            const int c = flat % CHUNK_K;
            ((float4*)ldsA)[i] =
                *(const float4*)(fk + (size_t)r * DDIM + kc + c);
        }
        __syncthreads();

        // prefetch next B chunk for this lane's rows
        if (kc + CHUNK_K < DDIM) {
            __builtin_prefetch(brow0 + kc + CHUNK_K, 0, 1);
            __builtin_prefetch(brow1 + kc + CHUNK_K, 0, 1);
        }

#pragma unroll
        for (int kk = 0; kk < CHUNK_K; kk += 4) {
            // B 4x16 fragments: lane holds B[2*half + 0..1][col]
            const v2f bf0 = *(const v2f*)(brow0 + kc + kk);
            const v2f bf1 = *(const v2f*)(brow1 + kc + kk);

            // Load ALL 8 A fragments first so the LDS loads batch up and the
            // 16 WMMAs below issue with only partial dscnt waits.
            v2f af[8];
#pragma unroll
            for (int t = 0; t < 8; ++t)
                af[t] = *(const v2f*)(&ldsA[(t * 16 + l16) * CHUNK_K + kk + 2 * half]);

#pragma unroll
            for (int t = 0; t < 8; ++t)
                acc0[t] = __builtin_amdgcn_wmma_f32_16x16x4_f32(
                    false, af[t], false, bf0, (short)0, acc0[t], false, false);
#pragma unroll
            for (int t = 0; t < 8; ++t)
                acc1[t] = __builtin_amdgcn_wmma_f32_16x16x4_f32(
                    false, af[t], false, bf1, (short)0, acc1[t], false, false);
        }
    }

    // C/D layout: VGPR i -> row (t*16 + i + 8*half), col n0 (+16) + l16
    const size_t base = (size_t)k * BDIM * NDIM;
#pragma unroll
    for (int t = 0; t < 8; ++t) {
        const int brw = t * 16 + 8 * half;
#pragma unroll
        for (int i = 0; i < 8; ++i) {
            const size_t off0 = base + (size_t)(brw + i) * NDIM + n0 + l16;
            sim[off0]       = acc0[t][i];
            soft[off0]      = 0.0f;
            sim[off0 + 16]  = acc1[t][i];
            soft[off0 + 16] = 0.0f;
        }
    }
}

// ---------------------------------------------------------------------------
// Kernel 3: per (k,b) row: top-5 of sim row, softmax(v/3) over those 5,
//           scatter into soft_labels (rest already zero).
// ---------------------------------------------------------------------------
__global__ void __launch_bounds__(256)
topk_softmax_kernel(const float* __restrict__ sim, float* __restrict__ soft) {
    const int row = blockIdx.x;                  // 0 .. K*B-1
    const int tid = threadIdx.x;                 // 0 .. 255
    const float* s = sim + (size_t)row * NDIM;

    // thread-local sorted top-5 (descending)
    float v[TOPK];
    int   c[TOPK];
#pragma unroll
    for (int i = 0; i < TOPK; ++i) { v[i] = -INFINITY; c[i] = -1; }

    for (int j = tid; j < NDIM; j += 256) {
        const float x = s[j];
        if (x > v[TOPK - 1]) {
            int p = TOPK - 1;
            while (p > 0 && x > v[p - 1]) { v[p] = v[p - 1]; c[p] = c[p - 1]; --p; }
            v[p] = x; c[p] = j;
        }
    }

    __shared__ float sv[256];
    __shared__ int   sc[256];
    __shared__ int   st[256];
    __shared__ float wv[TOPK];
    __shared__ int   wc[TOPK];

    int head = 0;
    for (int r = 0; r < TOPK; ++r) {
        sv[tid] = (head < TOPK) ? v[head] : -INFINITY;
        sc[tid] = (head < TOPK) ? c[head] : -1;
        st[tid] = tid;
        __syncthreads();
        for (int sft = 128; sft > 0; sft >>= 1) {
            if (tid < sft && sv[tid + sft] > sv[tid]) {
                sv[tid] = sv[tid + sft];
                sc[tid] = sc[tid + sft];
                st[tid] = st[tid + sft];
            }
            __syncthreads();
        }
        if (tid == 0) { wv[r] = sv[0]; wc[r] = sc[0]; }
        const int winner = st[0];
        __syncthreads();                  // protect sv/st reuse next round
        if (tid == winner) ++head;
    }

    if (tid < TOPK) {
        const float m = wv[0];
        float sum = 0.0f;
#pragma unroll
        for (int i = 0; i < TOPK; ++i) sum += __expf((wv[i] - m) * INV_TEMP);
        const float p = __expf((wv[tid] - m) * INV_TEMP) / sum;
        soft[(size_t)row * NDIM + wc[tid]] = p;
    }
}

// ---------------------------------------------------------------------------
extern "C" void kernel_launch(void* const* d_in, const int* in_sizes, int n_in,
                              void* d_out, int out_size, void* d_ws, size_t ws_size,
                              hipStream_t stream) {
    const float* part_features = (const float*)d_in[0];  // [K,B,D]
    const float* memory_bank   = (const float*)d_in[1];  // [K,N,D]

    float* soft = (float*)d_out;                          // [K,B,N]
    float* sim  = (float*)d_out + (size_t)KDIM * BDIM * NDIM;

    float* feats = (float*)d_ws;                          // [K,B,D] normalized

    // 1) normalize features
    normalize_kernel<<<KDIM * BDIM, DDIM, 0, stream>>>(part_features, feats);

    // 2) WMMA f32 GEMM: sim = feats @ memory_bank^T  (+ zero-fill soft)
    dim3 ggrid(NDIM / 128, KDIM, 1);
    sim_gemm_kernel<<<ggrid, 128, BDIM * CHUNK_K * sizeof(float), stream>>>(
        feats, memory_bank, sim, soft);

    // 3) top-5 + softmax scatter into soft_labels
    topk_softmax_kernel<<<KDIM * BDIM, 256, 0, stream>>>(sim, soft);
}